// Diffusion_16758962389776
// MI455X (gfx1250) — compile-verified
//
#include <hip/hip_runtime.h>

typedef __attribute__((ext_vector_type(2))) float v2f;
typedef __attribute__((ext_vector_type(8))) float v8f;
typedef __attribute__((ext_vector_type(4))) float f4v;
typedef __attribute__((ext_vector_type(4))) int   i4v;

#define TIMESTEPS 1000
#define BATCH     4
#define NDIM      2048
#define NROWS     (BATCH * NDIM)                   /* 8192 */
#define EDGES     (NDIM * (NDIM - 1) / 2)          /* 2,096,128 */
#define TOTALN    ((unsigned)(BATCH) * (unsigned)(EDGES))
#define LOG2_098  (-0.02914634686f)                /* log2(0.98) */

__device__ __forceinline__ float flip_at(int k) {
    // Qt[k] flip probability: 0.5 * (1 - 0.98^(k+1))
    return 0.5f * (1.0f - exp2f((float)(k + 1) * LOG2_098));
}

// BCE-with-logits term given logit qv and target probability qt.
__device__ __forceinline__ float bce_term(float qv, float qt) {
    return fmaxf(qv, 0.0f) - qv * qt + log1pf(expf(-fabsf(qv)));
}

__global__ __launch_bounds__(256)
void diffusion_loss_partial(const int*   __restrict__ adj,
                            const int*   __restrict__ tarr,
                            const float* __restrict__ u,
                            const float* __restrict__ qap,
                            float*       __restrict__ partial)
{
    const unsigned tid = threadIdx.x;
    // Two accumulators per thread -> fills the 16x4 f32 WMMA A-matrix
    // (2 VGPRs/lane * 32 lanes = 64 partials per wave).
    float acc0 = 0.0f, acc1 = 0.0f;

    // One row (b,i) = i contiguous lower-triangle elements.
    // b, i are block-uniform -> all Qt scalars live on the scalar unit.
    auto do_row = [&](unsigned row) {
        const unsigned b = row >> 11;              // row / NDIM
        const unsigned i = row & (NDIM - 1u);      // row % NDIM

        const int   tb = tarr[b];
        const float ft = flip_at(tb);
        const float fp = flip_at(tb == 0 ? TIMESTEPS - 1 : tb - 1);
        const float r_ft   = 1.0f / ft;
        const float r_1mft = 1.0f / (1.0f - ft);
        // q_target = Qt[0][x,1] * Qt[t-1][a0,1] / Qt[t][a0,x] : only 4 values.
        const float qt00 = 0.01f * fp          * r_1mft;   // a0=0, x=0
        const float qt01 = 0.99f * fp          * r_ft;     // a0=0, x=1
        const float qt10 = 0.01f * (1.0f - fp) * r_ft;     // a0=1, x=0
        const float qt11 = 0.99f * (1.0f - fp) * r_1mft;   // a0=1, x=1
        const float p1_0 = ft, p1_1 = 1.0f - ft;           // P(x=1 | a0)

        const unsigned base  = b * (unsigned)(NDIM * NDIM) + i * (unsigned)NDIM;
        const unsigned qbase = b * (unsigned)EDGES + i * (i - 1u) / 2u;
        const unsigned ifull = i & ~3u;            // full 4-wide groups

        // Body: every group fully valid -> no guards, no selects on loads.
        for (unsigned j0 = tid * 4u; j0 < ifull; j0 += 1024u) {
            // 16B-aligned, in-row b128 loads (j0+3 <= 2047 < NDIM).
            const i4v a4 = *(const i4v*)(adj + base + j0);
            const f4v u4 = *(const f4v*)(u   + base + j0);
            const float q0 = qap[qbase + j0 + 0u];   // in-bounds: j0+3 < i
            const float q1 = qap[qbase + j0 + 1u];
            const float q2 = qap[qbase + j0 + 2u];
            const float q3 = qap[qbase + j0 + 3u];
            #pragma unroll
            for (int k = 0; k < 4; ++k) {
                const int   a0 = a4[k];
                const float qv = (k == 0) ? q0 : (k == 1) ? q1 : (k == 2) ? q2 : q3;
                const float p1 = a0 ? p1_1 : p1_0;
                const bool  x  = u4[k] < p1;
                const float qt = a0 ? (x ? qt11 : qt10) : (x ? qt01 : qt00);
                const float term = bce_term(qv, qt);
                if (k & 1) acc1 += term; else acc0 += term;
            }
        }

        // Tail: 0-3 leftover elements, one (rotating) thread per row.
        if (tid == ((ifull >> 2u) & 255u)) {
            for (unsigned j = ifull; j < i; ++j) {
                const int   a0 = adj[base + j];
                const float uu = u[base + j];
                const float qv = qap[qbase + j];
                const float p1 = a0 ? p1_1 : p1_0;
                const bool  x  = uu < p1;
                const float qt = a0 ? (x ? qt11 : qt10) : (x ? qt01 : qt00);
                const float term = bce_term(qv, qt);
                if (j & 1u) acc1 += term; else acc0 += term;
            }
        }
    };

    // Pair rows (p, NROWS-1-p): each pair is exactly 2047 elements, so all
    // blocks do identical work (perfect static load balance, no tail blocks).
    for (unsigned p = blockIdx.x; p < (unsigned)(NROWS / 2); p += gridDim.x) {
        do_row(p);
        do_row((unsigned)NROWS - 1u - p);
    }

    // Wave reduction via V_WMMA_F32_16X16X4_F32 (exact, f32 in / f32 out):
    // A = the 64 wave partials (16x4), B = ones (4x16) => D[m,n] = rowsum_m(A).
    // Lane 0 VGPRs 0..7 hold D[0..7,0]; lane 16 VGPRs 0..7 hold D[8..15,0],
    // so (lane0 + lane16) lane-local sums give the full 64-way wave sum.
    // EXEC is all-ones here (no divergent returns above).
    v2f a;  a[0] = acc0; a[1] = acc1;
    v2f bo; bo[0] = 1.0f; bo[1] = 1.0f;
    v8f c = {};
    v8f d = __builtin_amdgcn_wmma_f32_16x16x4_f32(
        /*neg_a=*/false, a, /*neg_b=*/false, bo,
        /*c_mod=*/(short)0, c, /*reuse_a=*/false, /*reuse_b=*/false);
    const float lsum = d[0] + d[1] + d[2] + d[3] + d[4] + d[5] + d[6] + d[7];

    // Deterministic block combine (fixed-order summation, no float atomics).
    __shared__ float sred[16];                     // 8 waves * 2 halves
    const unsigned lane = tid & 31u;
    const unsigned wav  = tid >> 5u;
    if (lane == 0u)  sred[2u * wav]      = lsum;
    if (lane == 16u) sred[2u * wav + 1u] = lsum;
    __syncthreads();
    if (tid == 0u) {
        float s = 0.0f;
        #pragma unroll
        for (int k = 0; k < 16; ++k) s += sred[k];
        partial[blockIdx.x] = s;
    }
}

__global__ __launch_bounds__(256)
void diffusion_loss_final(const float* __restrict__ partial, int n,
                          float* __restrict__ out)
{
    __shared__ float sm[256];
    float s = 0.0f;
    for (int i = (int)threadIdx.x; i < n; i += 256) s += partial[i];
    sm[threadIdx.x] = s;
    __syncthreads();
    for (int off = 128; off > 0; off >>= 1) {
        if ((int)threadIdx.x < off) sm[threadIdx.x] += sm[threadIdx.x + off];
        __syncthreads();
    }
    if (threadIdx.x == 0u) out[0] = sm[0] * (1.0f / (float)TOTALN);
}

extern "C" void kernel_launch(void* const* d_in, const int* in_sizes, int n_in,
                              void* d_out, int out_size, void* d_ws, size_t ws_size,
                              hipStream_t stream)
{
    (void)in_sizes; (void)n_in; (void)out_size; (void)ws_size;
    const int*   adj = (const int*)  d_in[0];   // adj_start (B,N,N) int32
    const int*   t   = (const int*)  d_in[1];   // t (B,) int32
    const float* u   = (const float*)d_in[2];   // u (B,N,N) f32
    const float* qap = (const float*)d_in[3];   // q_approx (B,E) f32
    float* partial   = (float*)d_ws;            // 4 KB of scratch
    float* out       = (float*)d_out;           // scalar f32 loss

    const int NB = 1024;    // 4096 row-pairs / 1024 blocks = 4 pairs each,
                            // exactly 4*2047 elements per block.
    diffusion_loss_partial<<<NB, 256, 0, stream>>>(adj, t, u, qap, partial);
    diffusion_loss_final<<<1, 256, 0, stream>>>(partial, NB, out);
}